// Model_78460462564034
// MI455X (gfx1250) — compile-verified
//
#include <hip/hip_runtime.h>
#include <hip/hip_bf16.h>
#include <math.h>

#define NG        2304     // number of gaussians (48*48)
#define KK        25       // samples per gaussian (5*5)
#define GP_STRIDE 8        // packed gaussian stride (floats) = 32 B
#define IMG_STRIDE 32      // padded img row stride (floats)
#define HB_STRIDE 132      // LDS activation row stride; 132%64=4 -> conflict-free
#define TG        288      // gaussian tile for sampling (2304 = 8*288)
#define LOG2E     1.4426950408889634f
#define KEXP      (-0.7213475204444817f)   // -0.5 * log2(e)

typedef float v2f __attribute__((ext_vector_type(2)));
typedef float v8f __attribute__((ext_vector_type(8)));

__device__ __forceinline__ v8f wmma4(v2f a, v2f b, v8f c) {
    // D(16x16,f32) += A(16x4,f32) * B(4x16,f32) -- full f32 precision tensor op
    return __builtin_amdgcn_wmma_f32_16x16x4_f32(
        false, a, false, b, (short)0, c, false, false);
}

__device__ __forceinline__ float fast_exp2(float x) {
#if __has_builtin(__builtin_amdgcn_exp2f)
    return __builtin_amdgcn_exp2f(x);        // v_exp_f32
#else
    return exp2f(x);
#endif
}

__device__ __forceinline__ float fast_rcp(float x) {
#if __has_builtin(__builtin_amdgcn_rcpf)
    return __builtin_amdgcn_rcpf(x);         // v_rcp_f32
#else
    return 1.0f / x;
#endif
}

__device__ __forceinline__ float fast_tanh(float x) {
#if __has_builtin(__builtin_amdgcn_tanhf)
    return __builtin_amdgcn_tanhf(x);        // v_tanh_f32 (CDNA5 TRANS op)
#elif __has_builtin(__builtin_amdgcn_tanh_f32)
    return __builtin_amdgcn_tanh_f32(x);
#else
    return tanhf(x);
#endif
}

template <int ACT>
__device__ __forceinline__ float actf(float x) {
    if (ACT == 0) return fast_tanh(x);
    if (ACT == 1) return x * fast_rcp(1.0f + fast_exp2(-LOG2E * x));  // silu
    return x;
}

// ---------------------------------------------------------------- kernel 1
// Per-gaussian: conic = inv(cov), pack {mx,my,A,2B,C, opac*u, 0,0}.
__global__ void k_prep(const float* __restrict__ u, const float* __restrict__ means,
                       const float* __restrict__ scaling, const float* __restrict__ transform,
                       const float* __restrict__ opac, float* __restrict__ gpack) {
    int i = blockIdx.x * blockDim.x + threadIdx.x;
    if (i >= NG) return;
    float s0 = scaling[2 * i], s1 = scaling[2 * i + 1];
    float t  = transform[i];
    float c00 = s0 * s0, c01 = s0 * t, c11 = t * t + s1 * s1;
    float det = c00 * c11 - c01 * c01;
    float* g = gpack + (size_t)i * GP_STRIDE;
    g[0] = means[2 * i];
    g[1] = means[2 * i + 1];
    g[2] = c11 / det;           // A
    g[3] = -2.0f * c01 / det;   // 2B
    g[4] = c00 / det;           // C
    g[5] = opac[i] * u[i];      // weight
    g[6] = 0.0f;
    g[7] = 0.0f;
}

// ---------------------------------------------------------------- kernel 2
// All-pairs gaussian splat: 57600 samples x 2304 gaussians.
// 288 threads/block, 2 samples/thread, AoS 32B gaussian records in LDS.
__global__ void __launch_bounds__(TG)
k_sample(const float* __restrict__ means, const float* __restrict__ kern,
         const float* __restrict__ gpack, float* __restrict__ img) {
    __shared__ __align__(16) float gs[TG][GP_STRIDE];
    const int t  = blockIdx.x * TG + threadIdx.x;   // 0..28799 exact
    const int s0 = 2 * t, s1 = s0 + 1;
    const int i0 = s0 / KK, k0 = s0 - i0 * KK;
    const int i1 = s1 / KK, k1 = s1 - i1 * KK;
    const float sx0 = means[2 * i0]     + kern[2 * k0];
    const float sy0 = means[2 * i0 + 1] + kern[2 * k0 + 1];
    const float sx1 = means[2 * i1]     + kern[2 * k1];
    const float sy1 = means[2 * i1 + 1] + kern[2 * k1 + 1];
    float acc0 = 0.0f, acc1 = 0.0f;
    for (int base = 0; base < NG; base += TG) {
        const float4* gsrc = (const float4*)(gpack + (size_t)(base + threadIdx.x) * GP_STRIDE);
        float4 lo = gsrc[0];
        float4 hi = gsrc[1];
        __syncthreads();
        *(float4*)(&gs[threadIdx.x][0]) = lo;   // ds_store_b128 x2
        *(float4*)(&gs[threadIdx.x][4]) = hi;
        __syncthreads();
        if (base + TG < NG)   // pull next tile toward the WGP while computing
            __builtin_prefetch(gpack + (size_t)(base + TG + threadIdx.x) * GP_STRIDE, 0, 1);
#pragma unroll 4
        for (int j = 0; j < TG; ++j) {
            const float4 ga = *(const float4*)(&gs[j][0]);  // ds_load_b128 broadcast
            const float2 gb = *(const float2*)(&gs[j][4]);  // ds_load_b64  broadcast
            float dx0 = sx0 - ga.x, dy0 = sy0 - ga.y;
            float p0  = ga.z * dx0 * dx0 + ga.w * dx0 * dy0 + gb.x * dy0 * dy0;
            acc0 = fmaf(gb.y, fast_exp2(p0 * KEXP), acc0);
            float dx1 = sx1 - ga.x, dy1 = sy1 - ga.y;
            float p1  = ga.z * dx1 * dx1 + ga.w * dx1 * dy1 + gb.x * dy1 * dy1;
            acc1 = fmaf(gb.y, fast_exp2(p1 * KEXP), acc1);
        }
    }
    img[(size_t)i0 * IMG_STRIDE + k0] = acc0;
    img[(size_t)i1 * IMG_STRIDE + k1] = acc1;
}

// ---------------------------------------------------------------- kernel 3
// One FC layer for a wave's 16-row tile via v_wmma_f32_16x16x4_f32.
// A[m][k] = Asrc[m*a_stride + k] (k < kmax, else 0)
// B[k][n] = W[n*w_stride + k]    (n < nvalid, else 0) -- weights [out][in], x @ W^T
template <int ACT>
__device__ __forceinline__ void layer_fc(const float* __restrict__ Asrc, int a_stride,
                                         int kmax, int ksteps,
                                         const float* __restrict__ W, int w_stride, int nvalid,
                                         const float* __restrict__ bias, int nOut,
                                         float* __restrict__ dst, int colOff, int lane) {
    const int m    = lane & 15;   // A row / B column (per ISA 32-bit fragment layout)
    const int hi   = lane >> 4;
    const int koff = hi * 2;      // lanes 16-31 hold K halves +2,+3
    for (int nt = 0; nt < nOut / 16; ++nt) {
        const int  n   = nt * 16 + m;
        const bool nok = (n < nvalid);
        v8f c = {0.f, 0.f, 0.f, 0.f, 0.f, 0.f, 0.f, 0.f};
        for (int ks = 0; ks < ksteps; ++ks) {
            const int k0 = ks * 4 + koff;
            const int k1 = k0 + 1;
            v2f a, b;
            a.x = (k0 < kmax) ? Asrc[m * a_stride + k0] : 0.0f;
            a.y = (k1 < kmax) ? Asrc[m * a_stride + k1] : 0.0f;
            b.x = (nok && k0 < kmax) ? W[n * w_stride + k0] : 0.0f;
            b.y = (nok && k1 < kmax) ? W[n * w_stride + k1] : 0.0f;
            c = wmma4(a, b, c);
        }
        const float bcol = nok ? bias[n] : 0.0f;
#pragma unroll
        for (int r = 0; r < 8; ++r) {
            const int mr = r + 8 * hi;   // D VGPR r -> rows r / r+8
            dst[mr * HB_STRIDE + colOff + n] = actf<ACT>(c[r] + bcol);
        }
    }
}

// One full MLP pass (compile-time activation & output width -> no branches).
template <int ACT, int OD>
__device__ __forceinline__ void run_net(const float* __restrict__ img,
                                        const float* __restrict__ means, int g0,
                                        const float* __restrict__ cw, const float* __restrict__ cb,
                                        const float* __restrict__ pw, const float* __restrict__ pb,
                                        const float* __restrict__ w1, const float* __restrict__ v1,
                                        const float* __restrict__ w2, const float* __restrict__ v2,
                                        const float* __restrict__ w3, const float* __restrict__ v3,
                                        float* __restrict__ b0, float* __restrict__ b1,
                                        int lane, float* __restrict__ res) {
    // conv as (16,25)@(25,64) -> b0 cols 0..63 ; pe (16,2)@(2,32) -> b0 cols 64..95
    layer_fc<ACT>(img + (size_t)g0 * IMG_STRIDE, IMG_STRIDE, 25, 7, cw, 25, 64, cb, 64, b0, 0, lane);
    layer_fc<ACT>(means + (size_t)g0 * 2, 2, 2, 1, pw, 2, 32, pb, 32, b0, 64, lane);
    __syncthreads();
    layer_fc<ACT>(b0, HB_STRIDE, 96, 24, w1, 96, 128, v1, 128, b1, 0, lane);
    __syncthreads();
    layer_fc<ACT>(b1, HB_STRIDE, 128, 32, w2, 128, 128, v2, 128, b0, 0, lane);
    __syncthreads();
    // l3 as one WMMA N-tile (cols >= OD zero-padded), identity activation -> b1
    layer_fc<2>(b0, HB_STRIDE, 128, 32, w3, 128, OD, v3, 16, b1, 0, lane);
    __syncthreads();
    if (lane < 16) {
#pragma unroll
        for (int oc = 0; oc < OD; ++oc) res[oc] = b1[lane * HB_STRIDE + oc];
    }
    __syncthreads();
}

__global__ void __launch_bounds__(64)
k_net(const float* __restrict__ img, const float* __restrict__ means,
      const float* __restrict__ u, const float* __restrict__ scaling,
      const float* __restrict__ transform,
      const float* __restrict__ scw, const float* __restrict__ scb,
      const float* __restrict__ spw, const float* __restrict__ spb,
      const float* __restrict__ sw1, const float* __restrict__ sb1,
      const float* __restrict__ sw2, const float* __restrict__ sb2,
      const float* __restrict__ sw3, const float* __restrict__ sb3,
      const float* __restrict__ gcw, const float* __restrict__ gcb,
      const float* __restrict__ gpw, const float* __restrict__ gpb,
      const float* __restrict__ gw1, const float* __restrict__ gb1,
      const float* __restrict__ gw2, const float* __restrict__ gb2,
      const float* __restrict__ gw3, const float* __restrict__ gb3,
      float* __restrict__ out) {
    __shared__ float hb[2][2][16 * HB_STRIDE];   // [wave][pingpong], 33.8 KB
    const int lane = threadIdx.x & 31;
    const int wv   = threadIdx.x >> 5;
    const int g0   = blockIdx.x * 32 + wv * 16;  // wave's 16-gaussian tile
    float* b0 = hb[wv][0];
    float* b1 = hb[wv][1];

    float sres[1] = {0.f};
    float gz[6]   = {0.f, 0.f, 0.f, 0.f, 0.f, 0.f};

    // s-net: tanh activations, 1 output
    run_net<0, 1>(img, means, g0, scw, scb, spw, spb, sw1, sb1, sw2, sb2, sw3, sb3,
                  b0, b1, lane, sres);
    // g-net: silu activations, 6 outputs
    run_net<1, 6>(img, means, g0, gcw, gcb, gpw, gpb, gw1, gb1, gw2, gb2, gw3, gb3,
                  b0, b1, lane, gz);

    // Epilogue: combine + new conic, write (N,11) output row
    if (lane < 16) {
        const int i = g0 + lane;
        float un  = u[i] + sres[0];
        float mx  = means[2 * i]     + gz[0];
        float my  = means[2 * i + 1] + gz[1];
        float s0n = scaling[2 * i]     * expf(gz[2]);
        float s1n = scaling[2 * i + 1] * expf(gz[3]);
        float tn  = transform[i] + gz[4];
        float op  = 1.0f / (1.0f + expf(-gz[5]));
        float c00 = s0n * s0n, c01 = s0n * tn, c11 = tn * tn + s1n * s1n;
        float det = c00 * c11 - c01 * c01;
        float q0 =  c11 / det, q1 = -c01 / det, q3 = c00 / det;
        float* o = out + (size_t)i * 11;
        o[0] = un; o[1] = mx; o[2] = my; o[3] = s0n; o[4] = s1n;
        o[5] = tn; o[6] = op; o[7] = q0; o[8] = q1; o[9] = q1; o[10] = q3;
    }
}

// ---------------------------------------------------------------- launch
extern "C" void kernel_launch(void* const* d_in, const int* in_sizes, int n_in,
                              void* d_out, int out_size, void* d_ws, size_t ws_size,
                              hipStream_t stream) {
    const float* u         = (const float*)d_in[0];
    const float* means     = (const float*)d_in[1];
    const float* scaling   = (const float*)d_in[2];
    const float* transform = (const float*)d_in[3];
    const float* opac      = (const float*)d_in[4];
    const float* kern      = (const float*)d_in[5];

    float* ws    = (float*)d_ws;
    float* gpack = ws;                          // NG*8  floats
    float* img   = ws + (size_t)NG * GP_STRIDE; // NG*32 floats

    k_prep<<<(NG + 255) / 256, 256, 0, stream>>>(u, means, scaling, transform, opac, gpack);
    k_sample<<<(NG * KK) / (2 * TG), TG, 0, stream>>>(means, kern, gpack, img);
    k_net<<<NG / 32, 64, 0, stream>>>(
        img, means, u, scaling, transform,
        (const float*)d_in[6],  (const float*)d_in[7],  (const float*)d_in[8],  (const float*)d_in[9],
        (const float*)d_in[10], (const float*)d_in[11], (const float*)d_in[12], (const float*)d_in[13],
        (const float*)d_in[14], (const float*)d_in[15],
        (const float*)d_in[16], (const float*)d_in[17], (const float*)d_in[18], (const float*)d_in[19],
        (const float*)d_in[20], (const float*)d_in[21], (const float*)d_in[22], (const float*)d_in[23],
        (const float*)d_in[24], (const float*)d_in[25],
        (float*)d_out);
}